// lstm_20169166422798
// MI455X (gfx1250) — compile-verified
//
#include <hip/hip_runtime.h>
#include <math.h>

typedef float v2f __attribute__((ext_vector_type(2)));
typedef float v8f __attribute__((ext_vector_type(8)));

#define B_  64
#define T_  512
#define F_  513
#define H_  50
#define G_  200          // 4*H
#define M_  (B_ * T_)    // 32768
#define XG_STRIDE 208    // 200 padded to 13*16
#define HS_STRIDE 64     // 50 padded to 64

__device__ __forceinline__ float sigmoidf_(float v) {
    return 1.0f / (1.0f + expf(-v));
}

__device__ __forceinline__ int imin_(int a, int b) { return a < b ? a : b; }

// ---------------------------------------------------------------------------
// Kernel A: xg[m, n] = sum_k x[m,k] * W_ih[n,k] + b_ih[n]
// M=32768, K=513, N=200 (13 tiles of 16). One wave = 16 rows x full N.
// fp32 WMMA 16x16x4: A lane(row=lane&15) holds K = k + 2*(lane>=16) .. +1
//                    B lane(col=lane&15) holds K = k + 2*(lane>=16) .. +1
// Out-of-range N columns are loaded from a CLAMPED row pointer (their D
// columns are never stored), so the K-loop has no divergent branches.
// ---------------------------------------------------------------------------
__global__ __launch_bounds__(128)
void k_in_gemm(const float* __restrict__ x, const float* __restrict__ W_ih,
               const float* __restrict__ b_ih, float* __restrict__ xg)
{
    const int lane = threadIdx.x & 31;
    const int wave = threadIdx.x >> 5;
    const int mt   = blockIdx.x * 4 + wave;     // 2048 m-tiles
    const int m0   = mt * 16;
    const int row  = lane & 15;
    const int hi   = lane >> 4;                 // 0 or 1
    const int koff = 2 * hi;

    v8f acc[13] = {};
    const float* __restrict__ xrow = x + (size_t)(m0 + row) * F_ + koff;

    const float* wrowp[13];
    #pragma unroll
    for (int nt = 0; nt < 13; ++nt) {
        const int n = imin_(nt * 16 + row, G_ - 1);   // clamp: cols >=200 unused
        wrowp[nt] = W_ih + (size_t)n * F_ + koff;
    }

    for (int k = 0; k < 512; k += 4) {
        v2f a;
        a.x = xrow[k];
        a.y = xrow[k + 1];
        #pragma unroll
        for (int nt = 0; nt < 13; ++nt) {
            v2f b;
            b.x = wrowp[nt][k];
            b.y = wrowp[nt][k + 1];
            acc[nt] = __builtin_amdgcn_wmma_f32_16x16x4_f32(
                false, a, false, b, (short)0, acc[nt], false, false);
        }
    }
    // K tail: k = 512 (single valid column; k>=513 must contribute zero)
    {
        v2f a = {0.0f, 0.0f};
        if (hi == 0) a.x = xrow[512 - koff];
        #pragma unroll
        for (int nt = 0; nt < 13; ++nt) {
            v2f b = {0.0f, 0.0f};
            if (hi == 0) b.x = wrowp[nt][512 - koff];
            acc[nt] = __builtin_amdgcn_wmma_f32_16x16x4_f32(
                false, a, false, b, (short)0, acc[nt], false, false);
        }
    }
    // Store: C element r -> row m0 + r + 8*hi, col nt*16 + (lane&15)
    #pragma unroll
    for (int nt = 0; nt < 13; ++nt) {
        const int n = nt * 16 + row;
        if (nt < 12 || row < 8) {               // n < 200
            const float bias = b_ih[n];
            #pragma unroll
            for (int r = 0; r < 8; ++r) {
                const int m = m0 + r + 8 * hi;
                xg[(size_t)m * XG_STRIDE + n] = acc[nt][r] + bias;
            }
        }
    }
}

// ---------------------------------------------------------------------------
// Kernel B: recurrence over the batch axis. 512 independent sequences (one
// per t), 64 steps each. One wave per t; 8 waves / block share W_hh in LDS.
// ---------------------------------------------------------------------------
__global__ __launch_bounds__(256)
void k_recurrence(const float* __restrict__ xg, const float* __restrict__ W_hh,
                  const float* __restrict__ b_hh, float* __restrict__ hs)
{
    __shared__ float sW[G_ * H_];      // 200 x 50, 40 KB
    __shared__ float sb[G_];
    __shared__ float sh[8][64];        // h per wave (padded)
    __shared__ float sc[8][64];        // c per wave
    __shared__ float sg[8][G_];        // gate staging per wave

    for (int i = threadIdx.x; i < G_ * H_; i += 256) sW[i] = W_hh[i];
    for (int i = threadIdx.x; i < G_;      i += 256) sb[i] = b_hh[i];

    const int wave = threadIdx.x >> 5;
    const int lane = threadIdx.x & 31;
    const int t    = blockIdx.x * 8 + wave;

    sh[wave][lane]      = 0.0f;  sc[wave][lane]      = 0.0f;
    sh[wave][lane + 32] = 0.0f;  sc[wave][lane + 32] = 0.0f;
    __syncthreads();

    for (int b = 0; b < B_; ++b) {
        const float* __restrict__ grow = xg + (size_t)(b * T_ + t) * XG_STRIDE;
        // --- gate pre-activations: g[r] = xg[r] + b_hh[r] + W_hh[r,:] . h
        #pragma unroll
        for (int rr = 0; rr < 7; ++rr) {
            const int r = lane + rr * 32;
            if (r < G_) {
                float acc = grow[r] + sb[r];
                const float* __restrict__ w  = sW + r * H_;
                const float* __restrict__ hv = sh[wave];
                #pragma unroll
                for (int k = 0; k < H_; ++k) acc = fmaf(w[k], hv[k], acc);
                sg[wave][r] = acc;
            }
        }
        __syncthreads();
        // --- cell update, PyTorch gate order i,f,g,o
        float* __restrict__ hrow = hs + (size_t)(b * T_ + t) * HS_STRIDE;
        {
            const int j = lane;
            if (j < H_) {
                const float ig = sigmoidf_(sg[wave][j]);
                const float fg = sigmoidf_(sg[wave][j + H_]);
                const float gg = tanhf(sg[wave][j + 2 * H_]);
                const float og = sigmoidf_(sg[wave][j + 3 * H_]);
                const float c2 = fg * sc[wave][j] + ig * gg;
                const float h2 = og * tanhf(c2);
                sc[wave][j] = c2;
                sh[wave][j] = h2;
                hrow[j] = h2;
            }
        }
        {
            const int j = lane + 32;
            if (j < H_) {
                const float ig = sigmoidf_(sg[wave][j]);
                const float fg = sigmoidf_(sg[wave][j + H_]);
                const float gg = tanhf(sg[wave][j + 2 * H_]);
                const float og = sigmoidf_(sg[wave][j + 3 * H_]);
                const float c2 = fg * sc[wave][j] + ig * gg;
                const float h2 = og * tanhf(c2);
                sc[wave][j] = c2;
                sh[wave][j] = h2;
                hrow[j] = h2;
            } else {
                hrow[j] = 0.0f;                 // zero the 50..63 K-pad
            }
        }
        __syncthreads();
    }
}

// ---------------------------------------------------------------------------
// Kernel C: out[m, f] = sum_k hs[m,k] * W_out[f,k] + b_out[f]
// M=32768, K=50 (padded 64), N=513 (33 tiles, grouped 11 per wave).
// Same clamp-the-pointer trick for the n>=513 columns of the last tile.
// ---------------------------------------------------------------------------
__global__ __launch_bounds__(128)
void k_out_gemm(const float* __restrict__ hs, const float* __restrict__ W_out,
                const float* __restrict__ b_out, float* __restrict__ out)
{
    const int lane = threadIdx.x & 31;
    const int wave = threadIdx.x >> 5;
    const int flat = blockIdx.x * 4 + wave;     // 6144 waves
    const int ng   = flat >> 11;                // 0..2, n-tile group of 11
    const int mt   = flat & 2047;               // 2048 m-tiles
    const int m0   = mt * 16;
    const int row  = lane & 15;
    const int hi   = lane >> 4;
    const int koff = 2 * hi;

    v8f acc[11] = {};
    const float* __restrict__ arow = hs + (size_t)(m0 + row) * HS_STRIDE + koff;

    const float* wrowp[11];
    #pragma unroll
    for (int i = 0; i < 11; ++i) {
        const int n = imin_((ng * 11 + i) * 16 + row, F_ - 1); // clamp
        wrowp[i] = W_out + (size_t)n * H_ + koff;
    }

    for (int k = 0; k < 48; k += 4) {
        v2f a;
        a.x = arow[k];
        a.y = arow[k + 1];
        #pragma unroll
        for (int i = 0; i < 11; ++i) {
            v2f b;
            b.x = wrowp[i][k];
            b.y = wrowp[i][k + 1];
            acc[i] = __builtin_amdgcn_wmma_f32_16x16x4_f32(
                false, a, false, b, (short)0, acc[i], false, false);
        }
    }
    // K tail: k=48 -> lo lanes hold k=48,49; hi lanes (k=50,51) need zero B.
    // A hi lanes read the zeroed hs pad (harmless either way: B is zero).
    {
        v2f a;
        a.x = arow[48];
        a.y = arow[49];
        #pragma unroll
        for (int i = 0; i < 11; ++i) {
            v2f b = {0.0f, 0.0f};
            if (hi == 0) {
                b.x = wrowp[i][48];
                b.y = wrowp[i][49];
            }
            acc[i] = __builtin_amdgcn_wmma_f32_16x16x4_f32(
                false, a, false, b, (short)0, acc[i], false, false);
        }
    }
    #pragma unroll
    for (int i = 0; i < 11; ++i) {
        const int n = (ng * 11 + i) * 16 + row;
        if (n < F_) {
            const float bias = b_out[n];
            #pragma unroll
            for (int r = 0; r < 8; ++r) {
                const int m = m0 + r + 8 * hi;
                out[(size_t)m * F_ + n] = acc[i][r] + bias;
            }
        }
    }
}

// ---------------------------------------------------------------------------
extern "C" void kernel_launch(void* const* d_in, const int* in_sizes, int n_in,
                              void* d_out, int out_size, void* d_ws, size_t ws_size,
                              hipStream_t stream)
{
    (void)in_sizes; (void)n_in; (void)out_size; (void)ws_size;
    const float* x     = (const float*)d_in[0];
    const float* W_ih  = (const float*)d_in[1];
    const float* W_hh  = (const float*)d_in[2];
    const float* b_ih  = (const float*)d_in[3];
    const float* b_hh  = (const float*)d_in[4];
    const float* W_out = (const float*)d_in[5];
    const float* b_out = (const float*)d_in[6];
    float* out = (float*)d_out;

    // xg scratch lives in d_out (32768*208 floats << 32768*513); it is fully
    // consumed by kernel B before kernel C overwrites d_out with the result.
    float* xg = out;
    float* hs = (float*)d_ws;      // 32768 * 64 floats = 8.4 MB

    k_in_gemm   <<< 512, 128, 0, stream >>>(x, W_ih, b_ih, xg);
    k_recurrence<<<  64, 256, 0, stream >>>(xg, W_hh, b_hh, hs);
    k_out_gemm  <<<1536, 128, 0, stream >>>(hs, W_out, b_out, out);
}